// _SSMLayer_4209067950268
// MI455X (gfx1250) — compile-verified
//
#include <hip/hip_runtime.h>
#include <hip/hip_bf16.h>

// ---------------------------------------------------------------------------
// Diagonal linear SSM  s_t = sigmoid(A)*s_{t-1} + B*x_t ; y_t = C*s_t + bias
// Memory-bound (AI ~ 0.004 flop/byte; floor ~256MiB / 23.3TB/s ~= 11us).
// 3-phase chunked parallel scan; the two streaming passes use the CDNA5
// async global->LDS pipeline (ASYNCcnt) with a 2P-slot LDS ring so no DS
// drain is needed in the steady-state loop.
// ---------------------------------------------------------------------------

#ifndef PIPE
#define PIPE 8            // async loads in flight per block
#endif
#define SLOTS (2 * PIPE)  // LDS ring slots (reuse distance P -> no WAR hazard)

typedef __attribute__((ext_vector_type(4))) float v4f;  // native vector for NT stores

__device__ __forceinline__ float sigmoidf_fast(float v) {
    // v_rcp_f32 instead of the precise-division expansion; ~1ulp is fine here.
    return __builtin_amdgcn_rcpf(1.0f + __expf(-v));
}

template <int N>
__device__ __forceinline__ void wait_async_le() {
#if __has_builtin(__builtin_amdgcn_s_wait_asynccnt)
    __builtin_amdgcn_s_wait_asynccnt(N);
#else
    asm volatile("s_wait_asynccnt %0" :: "i"(N) : "memory");
#endif
}

// Per-lane async copy of 16B global -> LDS. GV addressing (64-bit vgpr pair).
// ISA: GLOBAL_LOAD_ASYNC_TO_LDS_B128, tracked with ASYNCcnt (in-order).
__device__ __forceinline__ void async_load_row_b128(unsigned lds_addr, const void* gaddr) {
    asm volatile("global_load_async_to_lds_b128 %0, %1, off"
                 :: "v"(lds_addr), "v"((unsigned long long)gaddr)
                 : "memory");
}

// Same, but last-use temporal hint: x is dead after this read (pass 3), so
// discard from cache instead of polluting L2 while y streams through it.
__device__ __forceinline__ void async_load_row_b128_lu(unsigned lds_addr, const void* gaddr) {
    asm volatile("global_load_async_to_lds_b128 %0, %1, off th:TH_LOAD_LU"
                 :: "v"(lds_addr), "v"((unsigned long long)gaddr)
                 : "memory");
}

// Flat pointers into the LDS aperture carry the LDS byte offset in bits[31:0]
// (CDNA5 ISA 10.2).
__device__ __forceinline__ unsigned lds_offset_of(const void* p) {
    return (unsigned)(unsigned long long)p;
}

__device__ __forceinline__ float4 sigmoid4(float4 v) {
    v.x = sigmoidf_fast(v.x); v.y = sigmoidf_fast(v.y);
    v.z = sigmoidf_fast(v.z); v.w = sigmoidf_fast(v.w);
    return v;
}

// ---------------------------------------------------------------------------
// Pass 1: per (chunk, batch) block, run recurrence from zero state and emit
// the chunk summary  h = sum_i a^(L-1-i) * B * x_i   -> hsum[b][c][d]
// ---------------------------------------------------------------------------
template <int P>
__global__ void ssm_chunk_local(const float* __restrict__ x,
                                const float* __restrict__ Ap,
                                const float* __restrict__ Bp,
                                float* __restrict__ hsum,
                                int T, int D, int L, int nc) {
    extern __shared__ float smem[];              // 2P rows of D floats
    const int tid = threadIdx.x;                 // D/4 threads
    const int c   = blockIdx.x;
    const int b   = blockIdx.y;
    const int d0  = tid * 4;
    const long long t0 = (long long)c * L;

    const float4 a4  = sigmoid4(*(const float4*)(Ap + d0));
    const float4 bp4 = *(const float4*)(Bp + d0);

    const char*  gbase    = (const char*)(x + ((long long)b * T + t0) * D + d0);
    const long long rowB  = (long long)D * sizeof(float);
    const unsigned  lrowB = (unsigned)D * 4u;
    const unsigned  lbase = lds_offset_of(smem) + (unsigned)(tid * 16);

#pragma unroll
    for (int i = 0; i < P; ++i)                  // row j -> slot j % (2P)
        async_load_row_b128(lbase + (unsigned)i * lrowB, gbase + (long long)i * rowB);

    float4 s = make_float4(0.f, 0.f, 0.f, 0.f);
    int i = 0;
    for (; i < L - P; ++i) {
        wait_async_le<P - 1>();                  // in-order: row i has landed
        float4 v = *(const float4*)&smem[(size_t)(i % (2 * P)) * D + d0];
        s.x = __builtin_fmaf(a4.x, s.x, bp4.x * v.x);
        s.y = __builtin_fmaf(a4.y, s.y, bp4.y * v.y);
        s.z = __builtin_fmaf(a4.z, s.z, bp4.z * v.z);
        s.w = __builtin_fmaf(a4.w, s.w, bp4.w * v.w);
        // refill slot (i+P)%(2P): last read P iterations ago, already consumed
        async_load_row_b128(lbase + (unsigned)((i + P) % (2 * P)) * lrowB,
                            gbase + (long long)(i + P) * rowB);
    }
    wait_async_le<0>();                          // drain tail rows
    for (; i < L; ++i) {
        float4 v = *(const float4*)&smem[(size_t)(i % (2 * P)) * D + d0];
        s.x = __builtin_fmaf(a4.x, s.x, bp4.x * v.x);
        s.y = __builtin_fmaf(a4.y, s.y, bp4.y * v.y);
        s.z = __builtin_fmaf(a4.z, s.z, bp4.z * v.z);
        s.w = __builtin_fmaf(a4.w, s.w, bp4.w * v.w);
    }
    *(float4*)(hsum + ((long long)b * nc + c) * D + d0) = s;
}

// ---------------------------------------------------------------------------
// Phase 2 (tiny): sequential scan over chunk summaries.
//   sinit[b][c] = state entering chunk c ;  s' = a^L * s + h[c]
// ---------------------------------------------------------------------------
__global__ void ssm_chunk_scan(const float* __restrict__ Ap,
                               const float* __restrict__ hsum,
                               float* __restrict__ sinit,
                               int D, int nc, int L) {
    const int b  = blockIdx.x;
    const int d0 = threadIdx.x * 4;
    const float4 a4 = sigmoid4(*(const float4*)(Ap + d0));
    const float lf = (float)L;
    float4 aL;
    aL.x = __powf(a4.x, lf); aL.y = __powf(a4.y, lf);
    aL.z = __powf(a4.z, lf); aL.w = __powf(a4.w, lf);

    float4 s = make_float4(0.f, 0.f, 0.f, 0.f);
    for (int c = 0; c < nc; ++c) {
        const long long off = ((long long)b * nc + c) * D + d0;
        *(float4*)(sinit + off) = s;
        const float4 h = *(const float4*)(hsum + off);
        s.x = __builtin_fmaf(aL.x, s.x, h.x);
        s.y = __builtin_fmaf(aL.y, s.y, h.y);
        s.z = __builtin_fmaf(aL.z, s.z, h.z);
        s.w = __builtin_fmaf(aL.w, s.w, h.w);
    }
}

// ---------------------------------------------------------------------------
// Pass 3: rerun each chunk from its correct initial state, emit y = C*s+bias.
// Same async streaming; x read hits L2 (last-use hint), y stores are
// non-temporal so the output stream does not evict x from L2.
// ---------------------------------------------------------------------------
template <int P>
__global__ void ssm_chunk_emit(const float* __restrict__ x,
                               const float* __restrict__ Ap,
                               const float* __restrict__ Bp,
                               const float* __restrict__ Cp,
                               const float* __restrict__ biasp,
                               const float* __restrict__ sinit,
                               float* __restrict__ y,
                               int T, int D, int L, int nc) {
    extern __shared__ float smem[];
    const int tid = threadIdx.x;
    const int c   = blockIdx.x;
    const int b   = blockIdx.y;
    const int d0  = tid * 4;
    const long long t0 = (long long)c * L;

    const float4 a4  = sigmoid4(*(const float4*)(Ap + d0));
    const float4 bp4 = *(const float4*)(Bp + d0);
    const float4 c4  = *(const float4*)(Cp + d0);
    const float4 bi4 = *(const float4*)(biasp + d0);

    const char*  gbase    = (const char*)(x + ((long long)b * T + t0) * D + d0);
    const long long rowB  = (long long)D * sizeof(float);
    const unsigned  lrowB = (unsigned)D * 4u;
    const unsigned  lbase = lds_offset_of(smem) + (unsigned)(tid * 16);

    v4f* yrow = (v4f*)(y + ((long long)b * T + t0) * D + d0);
    const int ystride = D / 4;

#pragma unroll
    for (int i = 0; i < P; ++i)
        async_load_row_b128_lu(lbase + (unsigned)i * lrowB, gbase + (long long)i * rowB);

    float4 s = *(const float4*)(sinit + ((long long)b * nc + c) * D + d0);
    int i = 0;
    for (; i < L - P; ++i) {
        wait_async_le<P - 1>();
        float4 v = *(const float4*)&smem[(size_t)(i % (2 * P)) * D + d0];
        s.x = __builtin_fmaf(a4.x, s.x, bp4.x * v.x);
        s.y = __builtin_fmaf(a4.y, s.y, bp4.y * v.y);
        s.z = __builtin_fmaf(a4.z, s.z, bp4.z * v.z);
        s.w = __builtin_fmaf(a4.w, s.w, bp4.w * v.w);
        v4f o;
        o.x = __builtin_fmaf(c4.x, s.x, bi4.x);
        o.y = __builtin_fmaf(c4.y, s.y, bi4.y);
        o.z = __builtin_fmaf(c4.z, s.z, bi4.z);
        o.w = __builtin_fmaf(c4.w, s.w, bi4.w);
        __builtin_nontemporal_store(o, &yrow[(size_t)i * ystride]);
        async_load_row_b128_lu(lbase + (unsigned)((i + P) % (2 * P)) * lrowB,
                               gbase + (long long)(i + P) * rowB);
    }
    wait_async_le<0>();
    for (; i < L; ++i) {
        float4 v = *(const float4*)&smem[(size_t)(i % (2 * P)) * D + d0];
        s.x = __builtin_fmaf(a4.x, s.x, bp4.x * v.x);
        s.y = __builtin_fmaf(a4.y, s.y, bp4.y * v.y);
        s.z = __builtin_fmaf(a4.z, s.z, bp4.z * v.z);
        s.w = __builtin_fmaf(a4.w, s.w, bp4.w * v.w);
        v4f o;
        o.x = __builtin_fmaf(c4.x, s.x, bi4.x);
        o.y = __builtin_fmaf(c4.y, s.y, bi4.y);
        o.z = __builtin_fmaf(c4.z, s.z, bi4.z);
        o.w = __builtin_fmaf(c4.w, s.w, bi4.w);
        __builtin_nontemporal_store(o, &yrow[(size_t)i * ystride]);
    }
}

// Fallback if workspace is too small for even one chunk summary: fully
// sequential per (b, 4-channel) thread. Correct but slow; not expected to run.
__global__ void ssm_fullseq(const float* __restrict__ x,
                            const float* __restrict__ Ap,
                            const float* __restrict__ Bp,
                            const float* __restrict__ Cp,
                            const float* __restrict__ biasp,
                            float* __restrict__ y,
                            int T, int D) {
    const int b  = blockIdx.x;
    const int d0 = threadIdx.x * 4;
    const float4 a4  = sigmoid4(*(const float4*)(Ap + d0));
    const float4 bp4 = *(const float4*)(Bp + d0);
    const float4 c4  = *(const float4*)(Cp + d0);
    const float4 bi4 = *(const float4*)(biasp + d0);
    float4 s = make_float4(0.f, 0.f, 0.f, 0.f);
    for (int t = 0; t < T; ++t) {
        const long long off = ((long long)b * T + t) * D + d0;
        float4 v = *(const float4*)(x + off);
        s.x = __builtin_fmaf(a4.x, s.x, bp4.x * v.x);
        s.y = __builtin_fmaf(a4.y, s.y, bp4.y * v.y);
        s.z = __builtin_fmaf(a4.z, s.z, bp4.z * v.z);
        s.w = __builtin_fmaf(a4.w, s.w, bp4.w * v.w);
        float4 o;
        o.x = __builtin_fmaf(c4.x, s.x, bi4.x);
        o.y = __builtin_fmaf(c4.y, s.y, bi4.y);
        o.z = __builtin_fmaf(c4.z, s.z, bi4.z);
        o.w = __builtin_fmaf(c4.w, s.w, bi4.w);
        *(float4*)(y + off) = o;
    }
}

extern "C" void kernel_launch(void* const* d_in, const int* in_sizes, int n_in,
                              void* d_out, int out_size, void* d_ws, size_t ws_size,
                              hipStream_t stream) {
    const float* x    = (const float*)d_in[0];
    const float* A    = (const float*)d_in[1];
    const float* Bp   = (const float*)d_in[2];
    const float* Cp   = (const float*)d_in[3];
    const float* bias = (const float*)d_in[4];

    const int Bsz = 4;                                   // per reference
    const int D   = in_sizes[1];
    const long long tot = (long long)in_sizes[0];
    const int T   = (int)(tot / ((long long)D * Bsz));
    const int thr = D / 4;                               // 256 for D=1024

    const size_t per = (size_t)Bsz * D * sizeof(float);  // one chunk-row of summaries
    if (ws_size < 2 * per || T < PIPE) {
        ssm_fullseq<<<dim3(Bsz), dim3(thr), 0, stream>>>(x, A, Bp, Cp, bias,
                                                         (float*)d_out, T, D);
        return;
    }

    int nc = 128;                                        // target: 512 blocks
    while (nc > 1 && (2 * per * (size_t)nc > ws_size || (T % nc) != 0 || (T / nc) < PIPE))
        nc >>= 1;
    const int L = T / nc;

    float* hsum  = (float*)d_ws;
    float* sinit = hsum + (size_t)Bsz * nc * D;

    dim3 grid(nc, Bsz), blk(thr);
    const size_t shmem = (size_t)SLOTS * D * sizeof(float);  // 64 KB at D=1024

    ssm_chunk_local<PIPE><<<grid, blk, shmem, stream>>>(x, A, Bp, hsum, T, D, L, nc);
    ssm_chunk_scan<<<dim3(Bsz), blk, 0, stream>>>(A, hsum, sinit, D, nc, L);
    ssm_chunk_emit<PIPE><<<grid, blk, shmem, stream>>>(x, A, Bp, Cp, bias, sinit,
                                                       (float*)d_out, T, D, L, nc);
}